// ColRepeatCausalLinear_76811195122557
// MI455X (gfx1250) — compile-verified
//
#include <hip/hip_runtime.h>

// ---------------------------------------------------------------------------
// ColRepeatCausalLinear on MI455X (gfx1250, wave32)
//
// cache[t] = decay*cache[t-1] + x[t]  (per column, T=4096, B*D=8192 columns)
// out[t]   = weight[t]*cache[t] + bias[t]
//
// Memory-bound problem (>=256MB traffic ~ 11us @ 23.3TB/s). 3-phase chunked
// scan (~384MB): (1) per-chunk carries, (2) tiny cross-chunk prefix,
// (3) tile-local scan via V_WMMA_F32_16X16X4_F32 (f32 precision, 16 MACs/elem
// extra compute == free) fused with weight/bias apply.
// ---------------------------------------------------------------------------

typedef float v2f __attribute__((ext_vector_type(2)));
typedef float v8f __attribute__((ext_vector_type(8)));

#define Bc   8
#define Tc   4096
#define Dc   1024
#define COLS (Bc * Dc)     // 8192 independent columns
#define CT   256           // timesteps per chunk
#define NCH  (Tc / CT)     // 16 chunks
#define TILE 16

__device__ __forceinline__ float clip_decay(const float* dv) {
    // decay = clip(v, 0.9, 1.0) ** (1/DECAY_CONSTANT), DECAY_CONSTANT == 1.0
    return fminf(fmaxf(dv[0], 0.9f), 1.0f);
}

__device__ __forceinline__ float powi(float x, int n) {
    float p = 1.0f;
    for (int i = 0; i < n; ++i) p *= x;
    return p;
}

// ---------------------------------------------------------------------------
// Phase 1: per-(chunk, column) carry: cache value at chunk end with zero
// carry-in. Streams x exactly once. Block = 256 consecutive columns (1KB
// coalesced per step), 512 blocks = 4096 waves -> saturates HBM.
// ---------------------------------------------------------------------------
__global__ void carry_kernel(const float* __restrict__ x,
                             const float* __restrict__ dv,
                             float* __restrict__ carries) {
    const int chunk = blockIdx.x >> 5;                       // / (COLS/256)
    const int col   = ((blockIdx.x & 31) << 8) + threadIdx.x;
    const float decay = clip_decay(dv);

    const int b = col >> 10;      // / Dc
    const int d = col & 1023;     // % Dc
    const float* px = x + ((size_t)(b * Tc + chunk * CT)) * Dc + d;

    float c = 0.0f;
#pragma unroll 4
    for (int s = 0; s < CT; ++s)
        c = fmaf(decay, c, px[(size_t)s * Dc]);

    carries[chunk * COLS + col] = c;
}

// ---------------------------------------------------------------------------
// Phase 2: per-column prefix over the NCH chunk carries with factor decay^CT.
// carry_in[j] = cache value entering chunk j (carry_in[0] = 0).
// ---------------------------------------------------------------------------
__global__ void prefix_kernel(const float* __restrict__ dv,
                              const float* __restrict__ carries,
                              float* __restrict__ carry_in) {
    const int col = blockIdx.x * blockDim.x + threadIdx.x;   // 0..COLS-1
    const float decay = clip_decay(dv);

    float dct = decay;                       // decay^(2^8) = decay^256 == decay^CT
    for (int i = 0; i < 8; ++i) dct *= dct;

    float P = 0.0f;
    for (int j = 0; j < NCH; ++j) {
        carry_in[j * COLS + col] = P;
        P = fmaf(dct, P, carries[j * COLS + col]);
    }
}

// ---------------------------------------------------------------------------
// Phase 3: one wave per (chunk x 16-column group). Per 16x16 tile:
//   D[M,N] = sum_K L[M,K]*x[K,N] + decay^(M+1)*carry[N],  L[M,K]=decay^(M-K)|0
// computed as 4 chained v_wmma_f32_16x16x4_f32 with the carry folded into the
// C accumulator. New carry = D[15,:] (lane 16+N, VGPR 7) via wave32 shuffle.
// Fused epilogue: out = weight[t]*cache + bias[t].
//
// VGPR layouts per CDNA5 ISA (wave32):
//   A 16x4 f32 : lane M=l&15; VGPR0/1 = K = 2*(l>>4) + {0,1}
//   B 4x16 f32 : col  N=l&15; VGPR0/1 = row K = 2*(l>>4) + {0,1}
//   C/D 16x16  : VGPR r -> M = r + 8*(l>>4), N = l&15
// ---------------------------------------------------------------------------
__global__ void scan_wmma_kernel(const float* __restrict__ x,
                                 const float* __restrict__ wt,
                                 const float* __restrict__ bias,
                                 const float* __restrict__ dv,
                                 const float* __restrict__ carry_in,
                                 float* __restrict__ out) {
    const int lane  = threadIdx.x & 31;
    const int wid   = blockIdx.x * (blockDim.x >> 5) + (threadIdx.x >> 5);
    const int chunk = wid & (NCH - 1);           // 0..15
    const int cg    = wid >> 4;                  // 0..COLS/16-1
    const int c0    = cg << 4;
    const int b     = c0 >> 10;
    const int d0    = c0 & 1023;
    const int t0    = chunk * CT;
    const int half  = lane >> 4;                 // 0 | 1
    const int ln    = lane & 15;

    const float decay = clip_decay(dv);

    // A operand: lower-triangular decay matrix, split into 4 K-chunks of 4.
    const int M = ln;
    v2f a[4];
#pragma unroll
    for (int kk = 0; kk < 4; ++kk) {
        const int K0 = 4 * kk + 2 * half;
        a[kk].x = (K0     <= M) ? powi(decay, M - K0)     : 0.0f;
        a[kk].y = (K0 + 1 <= M) ? powi(decay, M - K0 - 1) : 0.0f;
    }

    // Carry-in factor per accumulator register: decay^(M_r + 1), M_r = r+8*half
    float f[8];
#pragma unroll
    for (int r = 0; r < 8; ++r) f[r] = powi(decay, r + 8 * half + 1);

    float carry = carry_in[chunk * COLS + c0 + ln];   // per-column carry (dup across halves)

    for (int tt = 0; tt < CT / TILE; ++tt) {
        const int t = t0 + tt * TILE;

        // Seed accumulator with carry contribution: C[M,N] = decay^(M+1)*carry[N]
        v8f acc;
#pragma unroll
        for (int r = 0; r < 8; ++r) acc[r] = f[r] * carry;

        // 4 chained K=4 f32 WMMAs: acc += L[:,4k:4k+4] @ x[4k:4k+4,:]
#pragma unroll
        for (int kk = 0; kk < 4; ++kk) {
            const float* px = x + ((size_t)(b * Tc + t + 4 * kk + 2 * half)) * Dc + d0 + ln;
            v2f bm;
            bm.x = px[0];            // row K = 4kk + 2*half
            bm.y = px[Dc];           // row K = 4kk + 2*half + 1
            acc = __builtin_amdgcn_wmma_f32_16x16x4_f32(
                /*neg_a=*/false, a[kk], /*neg_b=*/false, bm,
                /*c_mod=*/(short)0, acc, /*reuse_a=*/false, /*reuse_b=*/false);
        }

        // New carry = cache at tile row M=15 -> lanes 16..31, VGPR 7
        carry = __shfl(acc[7], 16 + ln, 32);

        // Fused epilogue: out = weight*cache + bias (rows rbase..rbase+7)
        const int rbase = t + 8 * half;
        const float4 w0 = *(const float4*)(wt + rbase);
        const float4 w1 = *(const float4*)(wt + rbase + 4);
        const float4 g0 = *(const float4*)(bias + rbase);
        const float4 g1 = *(const float4*)(bias + rbase + 4);
        const float wv[8] = {w0.x, w0.y, w0.z, w0.w, w1.x, w1.y, w1.z, w1.w};
        const float bv[8] = {g0.x, g0.y, g0.z, g0.w, g1.x, g1.y, g1.z, g1.w};

        float* po = out + ((size_t)(b * Tc + rbase)) * Dc + d0 + ln;
#pragma unroll
        for (int r = 0; r < 8; ++r)
            po[(size_t)r * Dc] = fmaf(wv[r], acc[r], bv[r]);
    }
}

// ---------------------------------------------------------------------------
extern "C" void kernel_launch(void* const* d_in, const int* in_sizes, int n_in,
                              void* d_out, int out_size, void* d_ws, size_t ws_size,
                              hipStream_t stream) {
    const float* x      = (const float*)d_in[0];  // [B,T,D] f32
    const float* weight = (const float*)d_in[1];  // [1,T]   f32
    const float* bias   = (const float*)d_in[2];  // [T]     f32
    const float* dv     = (const float*)d_in[3];  // [1]     f32
    float* out = (float*)d_out;

    float* carries  = (float*)d_ws;               // NCH*COLS f32 = 512KB
    float* carry_in = carries + NCH * COLS;       // NCH*COLS f32 = 512KB

    // Phase 1: 16 chunks x 32 col-blocks = 512 blocks x 256 threads
    carry_kernel<<<dim3(NCH * (COLS / 256)), dim3(256), 0, stream>>>(x, dv, carries);

    // Phase 2: 8192 threads
    prefix_kernel<<<dim3(COLS / 256), dim3(256), 0, stream>>>(dv, carries, carry_in);

    // Phase 3: 8192 waves, 8 waves/block -> 1024 blocks
    scan_wmma_kernel<<<dim3((NCH * (COLS / 16)) / 8), dim3(256), 0, stream>>>(
        x, weight, bias, dv, carry_in, out);
}